// MLA_LossFunction_58205396795888
// MI455X (gfx1250) — compile-verified
//
#include <hip/hip_runtime.h>
#include <hip/hip_bf16.h>

typedef __attribute__((ext_vector_type(2))) float v2f;
typedef __attribute__((ext_vector_type(8))) float v8f;

#define WAVES          8
#define ROWS_PER_WAVE  16
#define CHUNK_ROWS     (WAVES * ROWS_PER_WAVE)   // 128
#define CHUNKS         4
#define ROWS_PER_BLOCK (CHUNK_ROWS * CHUNKS)     // 512
#define DIMS           128
#define RPAD           64                         // relations padded to 4 N-tiles of 16
#define PITCH          132                        // LDS row pitch in floats (132 % 64 == 4 -> conflict-free A frags)

// LDS layout (float offsets)
#define OFF_BFRAG  0                               // 4 * 32 * 32 * 2 = 8192 floats (B fragments, WMMA layout)
#define OFF_MO     8192                            // 8 waves * 16 * 132 = 16896 floats
#define OFF_WSQ    (OFF_MO + WAVES*ROWS_PER_WAVE*PITCH)   // 128: per-row ||mo_norm||^2
#define OFF_WSUM   (OFF_WSQ + 128)                 // 128: per-row sum(mo_norm)
#define OFF_RSQ    (OFF_WSUM + 128)                // 64: per-relation ||e||^2
#define OFF_RSUM   (OFF_RSQ + 64)                  // 64: per-relation sum(e)
#define OFF_RED    (OFF_RSUM + 64)                 // 8: per-wave partials
#define LDS_FLOATS (OFF_RED + 8)
#define LDS_BYTES  (LDS_FLOATS * 4)                // 101920 B (< 320 KB WGP LDS)

#define EPS   1e-6f
#define DEPS2 (128.0f * 1e-12f)
#define IW    1.0e-4f

__global__ void mla_zero_acc(double* acc) { acc[0] = 0.0; }

__global__ void mla_finalize(const double* __restrict__ acc, float* __restrict__ out) {
    out[0] = (float)(acc[0] + (double)IW);   // + INCORRECT_W * 1.0
}

__global__ __launch_bounds__(256) void mla_loss_main(
        const float* __restrict__ mo, const float* __restrict__ tg,
        const float* __restrict__ re, double* __restrict__ acc, int R)
{
    extern __shared__ float smem[];
    const int tid  = threadIdx.x;
    const int lane = tid & 31;
    const int wave = tid >> 5;

    // ---------------- setup: B fragments in WMMA register layout + relation stats --------------
    // B frag element (k_local, n): lane = n + 16*(k_local>=2), vgpr = k_local & 1  (16x16x4 f32 B, 4x16)
    // bfrag float2 index: ((nt*32 + s)*32 + lane)  -> holds re[n][4s+2*(lane>>4)+{0,1}], n=(lane&15)+16*nt
    #pragma unroll 4
    for (int i = 0; i < 16; ++i) {
        int idx = tid + i * 256;                 // 0 .. 4095 float2 entries
        int l   = idx & 31;
        int snt = idx >> 5;
        int s   = snt & 31;
        int nt  = snt >> 5;
        int k0  = 4 * s + 2 * (l >> 4);
        int n   = (l & 15) + 16 * nt;
        float a0 = 0.f, a1 = 0.f;
        if (n < R) { a0 = re[n * DIMS + k0]; a1 = re[n * DIMS + k0 + 1]; }
        smem[OFF_BFRAG + idx * 2 + 0] = a0;
        smem[OFF_BFRAG + idx * 2 + 1] = a1;
    }
    if (tid < RPAD) {
        float sq = 0.f, su = 0.f;
        if (tid < R) {
            for (int k = 0; k < DIMS; ++k) { float v = re[tid * DIMS + k]; sq += v * v; su += v; }
        }
        smem[OFF_RSQ + tid]  = sq;
        smem[OFF_RSUM + tid] = su;
    }
    __syncthreads();

    float* moLds = &smem[OFF_MO + wave * (ROWS_PER_WAVE * PITCH)];
    float local = 0.f;

    #pragma unroll 1
    for (int c = 0; c < CHUNKS; ++c) {
        const long rowBase = (long)blockIdx.x * ROWS_PER_BLOCK + (long)c * CHUNK_ROWS
                           + (long)wave * ROWS_PER_WAVE;

        // ---------------- phase A: stream rows, normalize, correct term, stage A tile -------
        #pragma unroll 1
        for (int r = 0; r < ROWS_PER_WAVE; ++r) {
            const float4 m4 = ((const float4*)(mo + (rowBase + r) * DIMS))[lane];
            float psq = m4.x * m4.x + m4.y * m4.y + m4.z * m4.z + m4.w * m4.w;
            float psu = m4.x + m4.y + m4.z + m4.w;
            #pragma unroll
            for (int msk = 1; msk < 32; msk <<= 1) {
                psq += __shfl_xor(psq, msk, 32);
                psu += __shfl_xor(psu, msk, 32);
            }
            const float rnorm = 1.0f / fmaxf(sqrtf(psq), 1e-12f);
            float4 n4;
            n4.x = m4.x * rnorm; n4.y = m4.y * rnorm; n4.z = m4.z * rnorm; n4.w = m4.w * rnorm;
            ((float4*)(moLds + r * PITCH))[lane] = n4;   // 528B row pitch, 16B aligned

            const float4 t4 = ((const float4*)(tg + (rowBase + r) * DIMS))[lane];
            const float dx = n4.x - t4.x + EPS, dy = n4.y - t4.y + EPS;
            const float dz = n4.z - t4.z + EPS, dw = n4.w - t4.w + EPS;
            float pc = dx * dx + dy * dy + dz * dz + dw * dw;
            #pragma unroll
            for (int msk = 1; msk < 32; msk <<= 1) pc += __shfl_xor(pc, msk, 32);

            if (lane == r) {
                local += sqrtf(pc);                                 // CORRECT_W = 1
                smem[OFF_WSQ  + wave * 16 + r] = psq * rnorm * rnorm;
                smem[OFF_WSUM + wave * 16 + r] = psu * rnorm;
            }
        }
        // same-wave LDS is in-order: A tile + stats readable without a block barrier.

        // ---------------- phase B: dots = mo_norm(16x128) @ re^T(128x64) via f32 WMMA --------
        // A frag element (m,k_local): lane = m + 16*(k_local>=2), vgpr = k_local & 1
        const float* aBase = moLds + (lane & 15) * PITCH + 2 * (lane >> 4);
        const float* bBase = &smem[OFF_BFRAG] + lane * 2;
        v8f acc0 = {}, acc1 = {}, acc2 = {}, acc3 = {};
        #pragma unroll
        for (int s = 0; s < 32; ++s) {
            const v2f a = *(const v2f*)(aBase + s * 4);
            acc0 = __builtin_amdgcn_wmma_f32_16x16x4_f32(false, a, false,
                     *(const v2f*)(bBase + (0 * 32 + s) * 64), (short)0, acc0, false, false);
            acc1 = __builtin_amdgcn_wmma_f32_16x16x4_f32(false, a, false,
                     *(const v2f*)(bBase + (1 * 32 + s) * 64), (short)0, acc1, false, false);
            acc2 = __builtin_amdgcn_wmma_f32_16x16x4_f32(false, a, false,
                     *(const v2f*)(bBase + (2 * 32 + s) * 64), (short)0, acc2, false, false);
            acc3 = __builtin_amdgcn_wmma_f32_16x16x4_f32(false, a, false,
                     *(const v2f*)(bBase + (3 * 32 + s) * 64), (short)0, acc3, false, false);
        }

        // ---------------- epilogue: distances, masked max over relations --------------------
        // C/D layout: vgpr j, lanes 0-15 -> M=j, lanes 16-31 -> M=j+8; N = lane&15 (+16*nt)
        float msq[8], msu[8], rmax[8];
        const int mhi = (lane >> 4) * 8;
        #pragma unroll
        for (int j = 0; j < 8; ++j) {
            msq[j]  = smem[OFF_WSQ  + wave * 16 + mhi + j];
            msu[j]  = smem[OFF_WSUM + wave * 16 + mhi + j];
            rmax[j] = -1.0f;
        }
        #pragma unroll
        for (int nt = 0; nt < 4; ++nt) {
            const int   n     = (lane & 15) + 16 * nt;
            const float rsq   = smem[OFF_RSQ + n];
            const float rsu   = smem[OFF_RSUM + n];
            const bool  valid = (n < R);
            const v8f   av    = (nt == 0) ? acc0 : (nt == 1) ? acc1 : (nt == 2) ? acc2 : acc3;
            #pragma unroll
            for (int j = 0; j < 8; ++j) {
                const float sq   = msq[j] + rsq - 2.0f * av[j]
                                 + 2.0f * EPS * (msu[j] - rsu) + DEPS2;
                const float dist = sqrtf(fmaxf(sq, 0.0f));
                if (valid) rmax[j] = fmaxf(rmax[j], dist);
            }
        }
        #pragma unroll
        for (int j = 0; j < 8; ++j) {
            #pragma unroll
            for (int msk = 1; msk < 16; msk <<= 1)
                rmax[j] = fmaxf(rmax[j], __shfl_xor(rmax[j], msk, 32));
            if ((lane & 15) == j) local -= IW * rmax[j];   // -INCORRECT_W * maxdist
        }
    }

    // ---------------- block reduction -> f64 atomic ------------------------------------------
    #pragma unroll
    for (int msk = 1; msk < 32; msk <<= 1) local += __shfl_xor(local, msk, 32);
    if (lane == 0) smem[OFF_RED + wave] = local;
    __syncthreads();
    if (tid == 0) {
        float t = 0.f;
        for (int w = 0; w < WAVES; ++w) t += smem[OFF_RED + w];
        atomicAdd(acc, (double)t);
    }
}

// Scalar tail for N % 512 rows (unused for N=1M, kept for robustness). One wave per row.
__global__ void mla_loss_tail(const float* __restrict__ mo, const float* __restrict__ tg,
                              const float* __restrict__ re, double* __restrict__ acc,
                              long startRow, long totalRows, int R)
{
    const long row  = startRow + (long)blockIdx.x * (blockDim.x / 32) + (threadIdx.x >> 5);
    const int  lane = threadIdx.x & 31;
    if (row >= totalRows) return;

    const float4 m4 = ((const float4*)(mo + row * DIMS))[lane];
    float psq = m4.x * m4.x + m4.y * m4.y + m4.z * m4.z + m4.w * m4.w;
    float psu = m4.x + m4.y + m4.z + m4.w;
    for (int msk = 1; msk < 32; msk <<= 1) {
        psq += __shfl_xor(psq, msk, 32);
        psu += __shfl_xor(psu, msk, 32);
    }
    const float rnorm = 1.0f / fmaxf(sqrtf(psq), 1e-12f);
    float4 n4; n4.x = m4.x*rnorm; n4.y = m4.y*rnorm; n4.z = m4.z*rnorm; n4.w = m4.w*rnorm;
    const float4 t4 = ((const float4*)(tg + row * DIMS))[lane];
    float dx = n4.x - t4.x + EPS, dy = n4.y - t4.y + EPS, dz = n4.z - t4.z + EPS, dw = n4.w - t4.w + EPS;
    float pc = dx*dx + dy*dy + dz*dz + dw*dw;
    for (int msk = 1; msk < 32; msk <<= 1) pc += __shfl_xor(pc, msk, 32);
    const float corr = sqrtf(pc);
    const float msq = psq * rnorm * rnorm, msm = psu * rnorm;

    float mx = 0.0f;
    for (int r = 0; r < R; ++r) {
        const float4 e4 = ((const float4*)(re + r * DIMS))[lane];
        float pd = n4.x*e4.x + n4.y*e4.y + n4.z*e4.z + n4.w*e4.w;
        float es = e4.x*e4.x + e4.y*e4.y + e4.z*e4.z + e4.w*e4.w;
        float eu = e4.x + e4.y + e4.z + e4.w;
        for (int msk = 1; msk < 32; msk <<= 1) {
            pd += __shfl_xor(pd, msk, 32);
            es += __shfl_xor(es, msk, 32);
            eu += __shfl_xor(eu, msk, 32);
        }
        const float sq = msq + es - 2.0f * pd + 2.0f * EPS * (msm - eu) + DEPS2;
        mx = fmaxf(mx, sqrtf(fmaxf(sq, 0.0f)));
    }
    if (lane == 0) atomicAdd(acc, (double)(corr - IW * mx));
}

extern "C" void kernel_launch(void* const* d_in, const int* in_sizes, int n_in,
                              void* d_out, int out_size, void* d_ws, size_t ws_size,
                              hipStream_t stream)
{
    const float* mo = (const float*)d_in[0];
    const float* tg = (const float*)d_in[1];
    const float* re = (const float*)d_in[2];
    double* acc = (double*)d_ws;
    float*  out = (float*)d_out;

    const long N = (long)in_sizes[0] / DIMS;       // 1,048,576 rows
    const int  R = in_sizes[2] / DIMS;             // 53 relations
    const long full = N / ROWS_PER_BLOCK;          // 2048 blocks of 512 rows
    const long rem  = N - full * ROWS_PER_BLOCK;

    mla_zero_acc<<<1, 1, 0, stream>>>(acc);
    if (full > 0)
        mla_loss_main<<<dim3((unsigned)full), dim3(256), LDS_BYTES, stream>>>(mo, tg, re, acc, R);
    if (rem > 0) {
        const long tb = (rem + 7) / 8;             // 8 rows (waves) per 256-thread block
        mla_loss_tail<<<dim3((unsigned)tb), dim3(256), 0, stream>>>(mo, tg, re, acc,
                                                                    full * ROWS_PER_BLOCK, N, R);
    }
    mla_finalize<<<1, 1, 0, stream>>>(acc, out);
}